// GCNVehicleTraffic_31971736551731
// MI455X (gfx1250) — compile-verified
//
#include <hip/hip_runtime.h>
#include <hip/hip_bf16.h>

// ---------------------------------------------------------------------------
// GCN (3x GCNConv + ELU, mean pool, linear 128->24) for MI455X / gfx1250.
// Bandwidth-bound: ~2.3GB traffic -> ~100us @ 23.3TB/s. GEMM (9.8 GFLOP) done
// with V_WMMA_F32_16X16X4_F32 (full f32 precision, wave computes 16x128 slab).
// Scatter-add hits the 192MB L2 (agg buffer = 51MB) via global_atomic_add_f32.
// ---------------------------------------------------------------------------

typedef float v2f __attribute__((ext_vector_type(2)));
typedef float v8f __attribute__((ext_vector_type(8)));

#define FEAT 128

__global__ void gcn_fill_f32(float* __restrict__ p, float v, int n) {
  int i = blockIdx.x * blockDim.x + threadIdx.x;
  if (i < n) p[i] = v;
}

// deg[c] += 1 for each edge target; buffer pre-filled with 1.0 (self loop).
__global__ void gcn_deg_count(const int* __restrict__ col, float* __restrict__ deg, int E) {
  int e = blockIdx.x * blockDim.x + threadIdx.x;
  if (e < E) unsafeAtomicAdd(&deg[col[e]], 1.0f);
}

__global__ void gcn_deg_rsqrt(float* __restrict__ d, int n) {
  int i = blockIdx.x * blockDim.x + threadIdx.x;
  if (i < n) d[i] = rsqrtf(d[i]);   // deg >= 1 always (self loops)
}

// Wt[n*128+k] = W[k*128+n]   (128x128, 16384 elements)
__global__ void gcn_transpose128(const float* __restrict__ W, float* __restrict__ Wt) {
  int idx = blockIdx.x * blockDim.x + threadIdx.x;
  int k = idx >> 7, n = idx & 127;
  Wt[(size_t)n * FEAT + k] = W[idx];
}

// H[N x 128] = X[N x 128] @ W[128 x 128] using V_WMMA_F32_16X16X4_F32.
// One wave32 computes a 16-row x 128-col slab: 8 accumulator tiles (v8f each),
// A fragment loaded once per k-step and reused across all 8 N-tiles.
// Fragment layouts (ISA 7.12.2): lane l, m = l&15, h = l>>4:
//   A (16x4):  VGPRj = A[m][2h + j]   -> contiguous v2f from row m
//   B (4x16):  VGPRj = B[2h + j][n]   -> contiguous v2f from Wt row n
//   C (16x16): VGPRj = C[j + 8h][n]
__global__ void gcn_gemm_wmma(const float* __restrict__ X, const float* __restrict__ Wt,
                              float* __restrict__ H, int nrows) {
  int lane = threadIdx.x & 31;
  int wave = blockIdx.x * (blockDim.x >> 5) + (threadIdx.x >> 5);
  int row0 = wave * 16;
  if (row0 >= nrows) return;                       // wave-uniform guard (EXEC all 1s)
  int m = lane & 15;
  int h = lane >> 4;

  const float* xrow = X + (size_t)(row0 + m) * FEAT;
  v8f z = {0.f, 0.f, 0.f, 0.f, 0.f, 0.f, 0.f, 0.f};
  v8f acc0 = z, acc1 = z, acc2 = z, acc3 = z, acc4 = z, acc5 = z, acc6 = z, acc7 = z;

#pragma unroll 8
  for (int k0 = 0; k0 < FEAT; k0 += 4) {
    v2f a = *(const v2f*)(xrow + k0 + 2 * h);
    const float* wb = Wt + (size_t)m * FEAT + k0 + 2 * h;   // Wt[(t*16+m)*128 + ...]
    v2f b0 = *(const v2f*)(wb + (size_t)0  * 16 * FEAT);
    v2f b1 = *(const v2f*)(wb + (size_t)1  * 16 * FEAT);
    v2f b2 = *(const v2f*)(wb + (size_t)2  * 16 * FEAT);
    v2f b3 = *(const v2f*)(wb + (size_t)3  * 16 * FEAT);
    v2f b4 = *(const v2f*)(wb + (size_t)4  * 16 * FEAT);
    v2f b5 = *(const v2f*)(wb + (size_t)5  * 16 * FEAT);
    v2f b6 = *(const v2f*)(wb + (size_t)6  * 16 * FEAT);
    v2f b7 = *(const v2f*)(wb + (size_t)7  * 16 * FEAT);
    acc0 = __builtin_amdgcn_wmma_f32_16x16x4_f32(false, a, false, b0, (short)0, acc0, false, false);
    acc1 = __builtin_amdgcn_wmma_f32_16x16x4_f32(false, a, false, b1, (short)0, acc1, false, false);
    acc2 = __builtin_amdgcn_wmma_f32_16x16x4_f32(false, a, false, b2, (short)0, acc2, false, false);
    acc3 = __builtin_amdgcn_wmma_f32_16x16x4_f32(false, a, false, b3, (short)0, acc3, false, false);
    acc4 = __builtin_amdgcn_wmma_f32_16x16x4_f32(false, a, false, b4, (short)0, acc4, false, false);
    acc5 = __builtin_amdgcn_wmma_f32_16x16x4_f32(false, a, false, b5, (short)0, acc5, false, false);
    acc6 = __builtin_amdgcn_wmma_f32_16x16x4_f32(false, a, false, b6, (short)0, acc6, false, false);
    acc7 = __builtin_amdgcn_wmma_f32_16x16x4_f32(false, a, false, b7, (short)0, acc7, false, false);
  }

  // Store: VGPR j of tile t -> H[row0 + j + 8h][t*16 + m]
  v8f* accs[8] = {&acc0, &acc1, &acc2, &acc3, &acc4, &acc5, &acc6, &acc7};
#pragma unroll
  for (int t = 0; t < 8; ++t) {
    v8f a = *accs[t];
#pragma unroll
    for (int j = 0; j < 8; ++j) {
      H[(size_t)(row0 + j + 8 * h) * FEAT + t * 16 + m] = a[j];
    }
  }
}

// agg[col[e]] += H[row[e]] * dinv[row]*dinv[col].  32 threads per edge, 4 floats each.
__global__ void gcn_scatter(const int* __restrict__ row, const int* __restrict__ col,
                            const float* __restrict__ dinv, const float* __restrict__ H,
                            float* __restrict__ agg, int E) {
  unsigned gid = blockIdx.x * blockDim.x + threadIdx.x;
  int e = (int)(gid >> 5);
  int g = (int)(gid & 31);
  if (e >= E) return;
  int r = row[e], c = col[e];
  float nrm = dinv[r] * dinv[c];
  float4 hv = *(const float4*)(H + (size_t)r * FEAT + g * 4);
  float* dst = agg + (size_t)c * FEAT + g * 4;
  unsafeAtomicAdd(dst + 0, hv.x * nrm);
  unsafeAtomicAdd(dst + 1, hv.y * nrm);
  unsafeAtomicAdd(dst + 2, hv.z * nrm);
  unsafeAtomicAdd(dst + 3, hv.w * nrm);
}

// In-place: agg = elu(agg + H*dinv^2 (self loop) + bias)
__global__ void gcn_epilogue(const float* __restrict__ H, const float* __restrict__ bias,
                             const float* __restrict__ dinv, float* __restrict__ agg, int total) {
  int idx = blockIdx.x * blockDim.x + threadIdx.x;
  if (idx >= total) return;
  int n = idx >> 7;
  int f = idx & 127;
  float di = dinv[n];
  float v = agg[idx] + H[idx] * di * di + bias[f];
  agg[idx] = (v > 0.f) ? v : (expf(v) - 1.f);
}

// Per-feature partial sums: block covers 128 nodes, thread f accumulates feature f.
__global__ void gcn_pool(const float* __restrict__ X, float* __restrict__ pooled, int N) {
  int f = threadIdx.x;                 // 0..127
  int n0 = blockIdx.x * 128;
  float s = 0.f;
  for (int i = 0; i < 128; ++i) {
    int n = n0 + i;
    if (n < N) s += X[(size_t)n * FEAT + f];
  }
  unsafeAtomicAdd(&pooled[f], s);
}

__global__ void gcn_final(const float* __restrict__ pooled, const float* __restrict__ lw,
                          const float* __restrict__ lb, float* __restrict__ out, float invN) {
  int j = threadIdx.x;
  if (j >= 24) return;
  float s = lb[j];
  for (int f = 0; f < FEAT; ++f) s += (pooled[f] * invN) * lw[f * 24 + j];
  out[j] = s;
}

extern "C" void kernel_launch(void* const* d_in, const int* in_sizes, int n_in,
                              void* d_out, int out_size, void* d_ws, size_t ws_size,
                              hipStream_t stream) {
  const float* x  = (const float*)d_in[0];
  const int*   ei = (const int*)d_in[1];
  const float* W[3] = {(const float*)d_in[2], (const float*)d_in[4], (const float*)d_in[6]};
  const float* b[3] = {(const float*)d_in[3], (const float*)d_in[5], (const float*)d_in[7]};
  const float* lw = (const float*)d_in[8];
  const float* lb = (const float*)d_in[9];
  float* out = (float*)d_out;

  int N = in_sizes[0] / FEAT;      // 100000 (multiple of 16)
  int E = in_sizes[1] / 2;         // 600000
  const int* row = ei;             // edge_index[0]: sources (gather)
  const int* col = ei + E;         // edge_index[1]: targets (scatter / degree)

  // Workspace layout
  char* ws = (char*)d_ws;
  size_t off = 0;
  auto wsalloc = [&](size_t bytes) -> void* {
    void* p = ws + off;
    off += (bytes + 255) & ~(size_t)255;
    return p;
  };
  float* dinv   = (float*)wsalloc((size_t)N * 4);
  float* Wt     = (float*)wsalloc((size_t)FEAT * FEAT * 4);
  float* P      = (float*)wsalloc((size_t)N * FEAT * 4);   // GEMM output h
  float* Q      = (float*)wsalloc((size_t)N * FEAT * 4);   // agg / x ping
  float* R      = (float*)wsalloc((size_t)N * FEAT * 4);   // agg / x pong
  float* pooled = (float*)wsalloc(128 * 4);
  (void)ws_size; (void)n_in; (void)out_size;

  const int T = 256;
  int total = N * FEAT;

  // Degrees: start at 1.0 (self loop), count targets, rsqrt in place.
  gcn_fill_f32<<<(N + T - 1) / T, T, 0, stream>>>(dinv, 1.0f, N);
  gcn_deg_count<<<(E + T - 1) / T, T, 0, stream>>>(col, dinv, E);
  gcn_deg_rsqrt<<<(N + T - 1) / T, T, 0, stream>>>(dinv, N);

  // Layer buffer rotation: L0: X=x   -> AGG=Q
  //                        L1: X=Q   -> AGG=R
  //                        L2: X=R   -> AGG=Q
  const float* Xin = x;
  float* aggbuf[3] = {Q, R, Q};

  int waves = N / 16;                                  // 6250 slabs of 16 rows
  int gemmBlocks = (waves + 7) / 8;                    // 8 waves per 256-thread block
  unsigned sthreads = (unsigned)E * 32u;

  for (int l = 0; l < 3; ++l) {
    float* AGG = aggbuf[l];
    gcn_transpose128<<<(FEAT * FEAT) / T, T, 0, stream>>>(W[l], Wt);
    gcn_gemm_wmma<<<gemmBlocks, 256, 0, stream>>>(Xin, Wt, P, N);
    gcn_fill_f32<<<(total + T - 1) / T, T, 0, stream>>>(AGG, 0.f, total);
    gcn_scatter<<<(int)((sthreads + T - 1) / T), T, 0, stream>>>(row, col, dinv, P, AGG, E);
    gcn_epilogue<<<(total + T - 1) / T, T, 0, stream>>>(P, b[l], dinv, AGG, total);
    Xin = AGG;
  }

  // Mean pool + final linear
  gcn_fill_f32<<<1, 128, 0, stream>>>(pooled, 0.f, 128);
  gcn_pool<<<(N + 127) / 128, 128, 0, stream>>>(Xin, pooled, N);
  gcn_final<<<1, 32, 0, stream>>>(pooled, lw, lb, out, 1.0f / (float)N);
}